// AdjEmbeddings_56899726737781
// MI455X (gfx1250) — compile-verified
//
#include <hip/hip_runtime.h>

typedef __attribute__((ext_vector_type(2))) float v2f;
typedef __attribute__((ext_vector_type(8))) float v8f;

#define DIM 128
#define K2  (2 * DIM)   // 256

// ---------------------------------------------------------------------------
// Kernel A: stream edge_src (12.8 MB, the roofline term), gather poi rows only
// for matched edges, produce per-block partial num[128] (f32, fixed serial
// order -> deterministic) and partial denom (int32 -> exact/deterministic).
// ---------------------------------------------------------------------------
__global__ void adj_scan_kernel(const int* __restrict__ user_idx,
                                const int* __restrict__ edge_src,
                                const int* __restrict__ edge_dst,
                                const int* __restrict__ edge_freq,
                                const float* __restrict__ poi,   // [P, DIM]
                                float* __restrict__ ws_num,      // [NB, DIM]
                                int*   __restrict__ ws_den,      // [NB]
                                int E, int NB)
{
    __shared__ float s_num[DIM];
    __shared__ int   s_dst[1024];
    __shared__ int   s_frq[1024];
    __shared__ int   s_flag;
    __shared__ int   s_red[256];

    const int tid = threadIdx.x;
    const int uid = user_idx[0];

    if (tid < DIM) s_num[tid] = 0.0f;
    if (tid == 0)  s_flag = 0;
    int den_loc = 0;
    __syncthreads();

    const int G = (E + 3) >> 2;                    // int4 groups
    const int gstride = gridDim.x * blockDim.x;    // groups per grid pass

    for (int base = blockIdx.x * blockDim.x; base < G; base += gstride) {
        const int g = base + tid;
        int md[4], mf[4];
        bool any = false;
        #pragma unroll
        for (int j = 0; j < 4; ++j) { md[j] = -1; mf[j] = 0; }

        if (g < G) {
            int srcs[4];
            if (g * 4 + 3 < E) {
                const int4 s4 = ((const int4*)edge_src)[g];
                srcs[0] = s4.x; srcs[1] = s4.y; srcs[2] = s4.z; srcs[3] = s4.w;
            } else {
                #pragma unroll
                for (int j = 0; j < 4; ++j) {
                    const int idx = g * 4 + j;
                    srcs[j] = (idx < E) ? edge_src[idx] : (uid ^ 1); // never match
                }
            }
            #pragma unroll
            for (int j = 0; j < 4; ++j) {
                const int idx = g * 4 + j;
                if (idx < E && srcs[j] == uid) {
                    md[j] = edge_dst[idx];
                    mf[j] = edge_freq[idx];
                    den_loc += mf[j];
                    any = true;
                }
            }
        }
        if (any) s_flag = 1;
        __syncthreads();
        const int flag = s_flag;       // uniform across block
        __syncthreads();               // all read flag before anyone re-writes

        if (flag) {                    // rare path (~1 block per launch)
            #pragma unroll
            for (int j = 0; j < 4; ++j) {
                s_dst[tid * 4 + j] = md[j];
                s_frq[tid * 4 + j] = mf[j];
            }
            __syncthreads();
            if (tid < DIM) {
                float acc = s_num[tid];
                for (int t = 0; t < 1024; ++t) {   // fixed order -> deterministic
                    const int d = s_dst[t];
                    if (d >= 0)
                        acc += (float)s_frq[t] * poi[(long)d * DIM + tid];
                }
                s_num[tid] = acc;
            }
            __syncthreads();
            if (tid == 0) s_flag = 0;
            __syncthreads();
        }
    }

    // exact integer block-reduce of denom
    s_red[tid] = den_loc;
    __syncthreads();
    for (int off = 128; off > 0; off >>= 1) {
        if (tid < off) s_red[tid] += s_red[tid + off];
        __syncthreads();
    }

    if (tid < DIM) ws_num[(size_t)blockIdx.x * DIM + tid] = s_num[tid];
    if (tid == 0)  ws_den[blockIdx.x] = s_red[0];
}

// ---------------------------------------------------------------------------
// Kernel B: 1 block, 8 waves. Fixed-order reduce of partials, build
// combined[256] = [u | num/denom], then FC via V_WMMA_F32_16X16X4_F32:
// wave w computes out[16w .. 16w+15]; A holds `combined` in row M=0 only.
// ---------------------------------------------------------------------------
__global__ void adj_fc_kernel(const int* __restrict__ user_idx,
                              const float* __restrict__ user_emb, // [U, DIM]
                              const float* __restrict__ fc_w,     // [DIM, 2*DIM]
                              const float* __restrict__ fc_b,     // [DIM]
                              const float* __restrict__ ws_num,   // [NB, DIM]
                              const int*   __restrict__ ws_den,   // [NB]
                              float* __restrict__ out,            // [DIM]
                              int NB)
{
    __shared__ float s_comb[K2];
    __shared__ int   s_red[256];

    const int tid = threadIdx.x;
    const int uid = user_idx[0];

    // reduce f32 partials in fixed b-order (deterministic), coalesced
    float num = 0.0f;
    if (tid < DIM) {
        #pragma unroll 4
        for (int b = 0; b < NB; ++b)
            num += ws_num[(size_t)b * DIM + tid];
    }
    // exact integer denom reduce
    int dl = 0;
    for (int b = tid; b < NB; b += 256) dl += ws_den[b];
    s_red[tid] = dl;
    __syncthreads();
    for (int off = 128; off > 0; off >>= 1) {
        if (tid < off) s_red[tid] += s_red[tid + off];
        __syncthreads();
    }
    const int denom = s_red[0];

    if (tid < DIM) {
        s_comb[tid]       = user_emb[(size_t)uid * DIM + tid];
        s_comb[DIM + tid] = (denom > 0) ? num / (float)denom : 0.0f;
    }
    __syncthreads();

    // ---- WMMA FC: D = A(16x4) x B(4x16) + C, chained over K=256 ----
    // 32-bit A 16x4 layout: lanes 0-15 -> {K=0,K=1}, lanes 16-31 -> {K=2,K=3}
    // 32-bit B 4x16 layout (transposed symmetry): lane half selects K pair.
    const int lane = tid & 31;
    const int wave = tid >> 5;
    const int n0   = wave * 16;          // output tile
    const int m    = lane & 15;
    const int kb   = (lane < 16) ? 0 : 2;

    v8f c = {};
    for (int k0 = 0; k0 < K2; k0 += 4) {
        v2f av, bv;
        av.x = (m == 0) ? s_comb[k0 + kb]     : 0.0f;  // A row 0 = combined
        av.y = (m == 0) ? s_comb[k0 + kb + 1] : 0.0f;
        bv.x = fc_w[(n0 + m) * K2 + (k0 + kb)];        // B[kk][nn] = fc_w[n0+nn][k0+kk]
        bv.y = fc_w[(n0 + m) * K2 + (k0 + kb + 1)];
        c = __builtin_amdgcn_wmma_f32_16x16x4_f32(
                /*neg_a=*/false, av, /*neg_b=*/false, bv,
                /*c_mod=*/(short)0, c, /*reuse_a=*/false, /*reuse_b=*/false);
    }
    // D row 0 lives in VGPR0 (c[0]) of lanes 0..15
    if (lane < 16)
        out[n0 + lane] = c[0] + fc_b[n0 + lane];
}

// ---------------------------------------------------------------------------
extern "C" void kernel_launch(void* const* d_in, const int* in_sizes, int n_in,
                              void* d_out, int out_size, void* d_ws, size_t ws_size,
                              hipStream_t stream) {
    const int*   user_idx  = (const int*)d_in[0];
    const float* poi       = (const float*)d_in[1];
    const int*   edge_src  = (const int*)d_in[2];
    const int*   edge_dst  = (const int*)d_in[3];
    const int*   edge_freq = (const int*)d_in[4];
    const float* user_emb  = (const float*)d_in[5];
    const float* fc_w      = (const float*)d_in[6];
    const float* fc_b      = (const float*)d_in[7];
    float*       out       = (float*)d_out;

    const int E = in_sizes[2];

    int NB = 512;  // enough blocks to saturate HBM on the 12.8 MB scan
    const size_t per_block = (size_t)DIM * 4 + 4;
    if ((size_t)NB * per_block > ws_size) {
        NB = (int)(ws_size / per_block);
        if (NB < 1) NB = 1;
    }

    float* ws_num = (float*)d_ws;
    int*   ws_den = (int*)((char*)d_ws + (size_t)NB * DIM * 4);

    adj_scan_kernel<<<NB, 256, 0, stream>>>(user_idx, edge_src, edge_dst,
                                            edge_freq, poi, ws_num, ws_den, E, NB);
    adj_fc_kernel<<<1, 256, 0, stream>>>(user_idx, user_emb, fc_w, fc_b,
                                         ws_num, ws_den, out, NB);
}